// q_phi1_58506044506600
// MI455X (gfx1250) — compile-verified
//
#include <hip/hip_runtime.h>
#include <hip/hip_bf16.h>

typedef float v2f __attribute__((ext_vector_type(2)));
typedef float v8f __attribute__((ext_vector_type(8)));

#define NNODES 16384
#define FEAT   256
#define NCLS   40
#define NCLSP  48
#define THETA  0.5f

__device__ __forceinline__ float fatomic_add(float* p, float v) {
    return __hip_atomic_fetch_add(p, v, __ATOMIC_RELAXED, __HIP_MEMORY_SCOPE_AGENT);
}

// ---------------- degree / dinv ----------------
__global__ void k_deg(const int* __restrict__ col, float* __restrict__ deg, int E) {
    int e = blockIdx.x * blockDim.x + threadIdx.x;
    if (e < E) fatomic_add(&deg[col[e]], 1.0f);
}

__global__ void k_dinv(float* __restrict__ deg, int n) {
    int i = blockIdx.x * blockDim.x + threadIdx.x;
    if (i < n) deg[i] = rsqrtf(deg[i] + 1.0f);  // +1 self-loop, always > 0
}

// ---------------- pack B [256 x srcCols] -> paired-K fragments [128 x dstCols] of float2 ----
// Bp[p*dstCols + c] = { B[2p][c], B[2p+1][c] }, zero-padded for c >= srcCols.
__global__ void k_packB(const float* __restrict__ B, v2f* __restrict__ Bp,
                        int srcCols, int dstCols) {
    int i = blockIdx.x * blockDim.x + threadIdx.x;
    if (i >= (FEAT / 2) * dstCols) return;
    int p = i / dstCols, c = i % dstCols;
    v2f v;
    v.x = (c < srcCols) ? B[(2 * p) * srcCols + c] : 0.0f;
    v.y = (c < srcCols) ? B[(2 * p + 1) * srcCols + c] : 0.0f;
    Bp[i] = v;
}

// ---------------- fp32 WMMA GEMM: C[M x ldc] = A[M x 256] * B[256 x ldb] (+bias) ----------------
// B pre-packed into paired-K float2 layout (ldb = packed column count).
// One wave computes a 16 x (16*NT) tile; K stepped by 4; software-pipelined fragment loads.
template <int NT>
__global__ __launch_bounds__(256) void k_gemm_wmma(
        const float* __restrict__ A, const v2f* __restrict__ Bp,
        float* __restrict__ C, int M, int ldb, int ldc,
        int maxCol, int colGroups, const float* __restrict__ bias) {
    int w = blockIdx.x * (blockDim.x >> 5) + (threadIdx.x >> 5);
    int rowTile  = w / colGroups;
    int colGroup = w % colGroups;
    if (rowTile * 16 >= M) return;
    int lane = threadIdx.x & 31;
    int half = lane >> 4;          // K-half selector for A/B fragments
    int lm   = lane & 15;          // M (for A) / N (for B,D) within tile
    int rowBase = rowTile * 16;
    int colBase = colGroup * (16 * NT);

    const float* arow  = A + (size_t)(rowBase + lm) * FEAT + 2 * half;
    const v2f*   bbase = Bp + (size_t)half * ldb + colBase + lm;

    v8f acc[NT] = {};

    // prologue: fragments for k0 = 0
    v2f a_cur = *(const v2f*)(arow);
    v2f b_cur[NT];
#pragma unroll
    for (int t = 0; t < NT; ++t) b_cur[t] = bbase[t * 16];

#pragma unroll 2
    for (int k0 = 0; k0 < FEAT - 4; k0 += 4) {
        // prefetch next iteration's fragments (in flight during the WMMAs below)
        v2f a_nxt = *(const v2f*)(arow + k0 + 4);
        const v2f* bn = bbase + (size_t)((k0 + 4) >> 1) * ldb;
        v2f b_nxt[NT];
#pragma unroll
        for (int t = 0; t < NT; ++t) b_nxt[t] = bn[t * 16];
#pragma unroll
        for (int t = 0; t < NT; ++t)
            acc[t] = __builtin_amdgcn_wmma_f32_16x16x4_f32(
                false, a_cur, false, b_cur[t], (short)0, acc[t], false, false);
        a_cur = a_nxt;
#pragma unroll
        for (int t = 0; t < NT; ++t) b_cur[t] = b_nxt[t];
    }
    // epilogue: last K step
#pragma unroll
    for (int t = 0; t < NT; ++t)
        acc[t] = __builtin_amdgcn_wmma_f32_16x16x4_f32(
            false, a_cur, false, b_cur[t], (short)0, acc[t], false, false);

#pragma unroll
    for (int t = 0; t < NT; ++t) {
        int c = colBase + t * 16 + lm;
        if (c < maxCol) {
            float bv = bias ? bias[c] : 0.0f;
#pragma unroll
            for (int v = 0; v < 8; ++v)
                C[(size_t)(rowBase + v + 8 * half) * ldc + c] = acc[t][v] + bv;
        }
    }
}

// ---------------- edge aggregation: agg[c] += dinv[r]*dinv[c] * xw[r] ----------------
// 64 threads per edge, float4 per thread, hardware f32 atomics.
__global__ void k_agg(const int* __restrict__ row, const int* __restrict__ col,
                      const float* __restrict__ dinv, const float* __restrict__ xw,
                      float* __restrict__ agg, int E) {
    int g = blockIdx.x * blockDim.x + threadIdx.x;
    int e = g >> 6;
    int t = g & 63;
    if (e >= E) return;
    int r = row[e], c = col[e];
    float nrm = dinv[r] * dinv[c];
    const float4 v = *(const float4*)(xw + (size_t)r * FEAT + t * 4);
    float* dst = agg + (size_t)c * FEAT + t * 4;
    fatomic_add(dst + 0, nrm * v.x);
    fatomic_add(dst + 1, nrm * v.y);
    fatomic_add(dst + 2, nrm * v.z);
    fatomic_add(dst + 3, nrm * v.w);
}

// ---------------- finalize: out = (agg + dinv^2 * xw + b), optional relu ----------------
__global__ void k_finalize(const float* __restrict__ agg, const float* __restrict__ xw,
                           const float* __restrict__ dinv, const float* __restrict__ bias,
                           float* __restrict__ out, int relu) {
    int i = blockIdx.x * blockDim.x + threadIdx.x;
    if (i >= NNODES * FEAT) return;
    int r = i >> 8;
    int c = i & (FEAT - 1);
    float d = dinv[r];
    float v = agg[i] + d * d * xw[i] + bias[c];
    out[i] = relu ? fmaxf(v, 0.0f) : v;
}

// ---------------- row L2 normalize (applied twice, as in reference) ----------------
__global__ void k_l2norm(const float* __restrict__ h, float* __restrict__ rep) {
    int w = blockIdx.x * (blockDim.x >> 5) + (threadIdx.x >> 5);
    if (w >= NNODES) return;
    int lane = threadIdx.x & 31;
    const float* src = h + (size_t)w * FEAT + lane * 8;
    float4 p0 = *(const float4*)(src);
    float4 p1 = *(const float4*)(src + 4);
    float ss = p0.x*p0.x + p0.y*p0.y + p0.z*p0.z + p0.w*p0.w
             + p1.x*p1.x + p1.y*p1.y + p1.z*p1.z + p1.w*p1.w;
#pragma unroll
    for (int off = 16; off >= 1; off >>= 1) ss += __shfl_xor(ss, off, 32);
    float s1  = 1.0f / fmaxf(sqrtf(ss), 1e-12f);
    float ss2 = ss * s1 * s1;                               // second normalize (idempotent)
    float s   = s1 * (1.0f / fmaxf(sqrtf(ss2), 1e-12f));
    float* dst = rep + (size_t)w * FEAT + lane * 8;
    *(float4*)(dst)     = make_float4(p0.x*s, p0.y*s, p0.z*s, p0.w*s);
    *(float4*)(dst + 4) = make_float4(p1.x*s, p1.y*s, p1.z*s, p1.w*s);
}

// ---------------- edge loss terms: wave per edge ----------------
// mode==1: pos edges (uses sim + target), mode==0: neg edges
__global__ void k_loss(const int* __restrict__ r, const int* __restrict__ c,
                       const float* __restrict__ rep, const float* __restrict__ sim,
                       const float* __restrict__ tgt, float* __restrict__ accum,
                       int E, int mode) {
    int w = blockIdx.x * (blockDim.x >> 5) + (threadIdx.x >> 5);
    if (w >= E) return;
    int rr = r[w], cc = c[w];
    if (rr >= cc) return;  // mask: contributes nothing to sse or count
    int lane = threadIdx.x & 31;
    const float* a = rep + (size_t)rr * FEAT + lane * 8;
    const float* b = rep + (size_t)cc * FEAT + lane * 8;
    float4 a0 = *(const float4*)a, a1 = *(const float4*)(a + 4);
    float4 b0 = *(const float4*)b, b1 = *(const float4*)(b + 4);
    float dp = a0.x*b0.x + a0.y*b0.y + a0.z*b0.z + a0.w*b0.w
             + a1.x*b1.x + a1.y*b1.y + a1.z*b1.z + a1.w*b1.w;
#pragma unroll
    for (int off = 16; off >= 1; off >>= 1) dp += __shfl_xor(dp, off, 32);
    if (lane == 0) {
        float wgt = fmaxf(dp, 0.0f);
        float term;
        if (mode) {
            float pos = sim[(size_t)rr * NNODES + cc] * THETA + wgt * (1.0f - THETA);
            float d = pos - tgt[0];
            term = d * d;
        } else {
            term = wgt * wgt;
        }
        fatomic_add(&accum[0], term);
        fatomic_add(&accum[1], 1.0f);
    }
}

__global__ void k_final(const float* __restrict__ accum, float* __restrict__ out_loss) {
    if (threadIdx.x == 0 && blockIdx.x == 0)
        out_loss[0] = accum[0] * (float)NNODES / accum[1];
}

extern "C" void kernel_launch(void* const* d_in, const int* in_sizes, int n_in,
                              void* d_out, int out_size, void* d_ws, size_t ws_size,
                              hipStream_t stream) {
    const int*   edge  = (const int*)d_in[0];     // [2,E]
    const int*   nedge = (const int*)d_in[1];     // [2,E]
    const float* feats = (const float*)d_in[2];   // [N,256]
    const float* sim   = (const float*)d_in[3];   // [N,N]
    const float* tgt   = (const float*)d_in[4];   // [1]
    const float* W1    = (const float*)d_in[5];
    const float* b1    = (const float*)d_in[6];
    const float* W2    = (const float*)d_in[7];
    const float* b2    = (const float*)d_in[8];
    const float* Wy    = (const float*)d_in[9];
    const float* by    = (const float*)d_in[10];
    const int E = in_sizes[0] / 2;

    // workspace layout
    float* buf0  = (float*)d_ws;                   // xw1 / xw2 / h2   (N*256)
    float* buf1  = buf0 + (size_t)NNODES * FEAT;   // agg1/h1, agg2    (N*256)
    float* dinv  = buf1 + (size_t)NNODES * FEAT;   // deg -> dinv      (N)
    v2f*   W1p   = (v2f*)(dinv + NNODES);          // packed W1 (128x256 v2f)
    v2f*   W2p   = W1p + (FEAT / 2) * FEAT;        // packed W2
    v2f*   Wyp   = W2p + (FEAT / 2) * FEAT;        // packed, padded Wy (128x48 v2f)
    float* accum = (float*)(Wyp + (FEAT / 2) * NCLSP);  // {sse, count}

    float* rep   = (float*)d_out;                          // [N,256]
    float* lossp = rep + (size_t)NNODES * FEAT;            // scalar
    float* yout  = lossp + 1;                              // [N,40]

    hipMemsetAsync(dinv,  0, NNODES * sizeof(float), stream);
    hipMemsetAsync(buf1,  0, (size_t)NNODES * FEAT * sizeof(float), stream);
    hipMemsetAsync(accum, 0, 2 * sizeof(float), stream);

    // degrees + dinv
    k_deg <<<(E + 255) / 256, 256, 0, stream>>>(edge + E, dinv, E);
    k_dinv<<<(NNODES + 255) / 256, 256, 0, stream>>>(dinv, NNODES);

    // pack weights into paired-K fragment layout (one b64 load per B fragment)
    k_packB<<<((FEAT / 2) * FEAT  + 255) / 256, 256, 0, stream>>>(W1, W1p, FEAT, FEAT);
    k_packB<<<((FEAT / 2) * FEAT  + 255) / 256, 256, 0, stream>>>(W2, W2p, FEAT, FEAT);
    k_packB<<<((FEAT / 2) * NCLSP + 255) / 256, 256, 0, stream>>>(Wy, Wyp, NCLS, NCLSP);

    // --- layer 1 ---
    // 1024 row tiles * 2 col groups (128 cols each) = 2048 waves, 8 waves/block
    k_gemm_wmma<8><<<256, 256, 0, stream>>>(feats, W1p, buf0, NNODES, FEAT, FEAT, FEAT, 2, nullptr);
    k_agg<<<(E * 64 + 255) / 256, 256, 0, stream>>>(edge, edge + E, dinv, buf0, buf1, E);
    k_finalize<<<(NNODES * FEAT + 255) / 256, 256, 0, stream>>>(buf1, buf0, dinv, b1, buf1, 1);

    // --- layer 2 ---
    k_gemm_wmma<8><<<256, 256, 0, stream>>>(buf1, W2p, buf0, NNODES, FEAT, FEAT, FEAT, 2, nullptr);
    hipMemsetAsync(buf1, 0, (size_t)NNODES * FEAT * sizeof(float), stream);
    k_agg<<<(E * 64 + 255) / 256, 256, 0, stream>>>(edge, edge + E, dinv, buf0, buf1, E);
    k_finalize<<<(NNODES * FEAT + 255) / 256, 256, 0, stream>>>(buf1, buf0, dinv, b2, buf0, 0);

    // rep = l2norm(l2norm(h2))
    k_l2norm<<<NNODES / 8, 256, 0, stream>>>(buf0, rep);

    // y = rep @ Wy + by  (packed/padded to 48 cols, stores guarded at 40)
    k_gemm_wmma<3><<<128, 256, 0, stream>>>(rep, Wyp, yout, NNODES, NCLSP, NCLS, NCLS, 1, by);

    // reconstruction loss (neg then pos), wave per edge (8 waves/block)
    k_loss<<<(E + 7) / 8, 256, 0, stream>>>(nedge, nedge + E, rep, nullptr, nullptr, accum, E, 0);
    k_loss<<<(E + 7) / 8, 256, 0, stream>>>(edge,  edge + E,  rep, sim,     tgt,     accum, E, 1);
    k_final<<<1, 32, 0, stream>>>(accum, lossp);
}